// Router_40699110096909
// MI455X (gfx1250) — compile-verified
//
#include <hip/hip_runtime.h>

// Router: logits = x @ W^T, softmax, argmax one-hot, top prob.
// x: [65536, 2048] f32, W: [64, 2048] f32.
// d_out = [one_hot int32 65536*64][top_prob f32 65536][logits f32 65536*64]

typedef float v2f __attribute__((ext_vector_type(2)));
typedef float v8f __attribute__((ext_vector_type(8)));

#define D_MODEL      2048
#define NUM_EXPERTS  64
#define T_PER_BLOCK  128
#define BK           64
#define LDST         68   // padded LDS row stride (floats): conflict-free b64 frag loads
#define XTILE        (T_PER_BLOCK * LDST)
#define WTILE        (NUM_EXPERTS * LDST)

// CDNA5 async global->LDS copy (ASYNCcnt-tracked, no VGPR round trip).
// GVS mode: mem_addr = SGPR64 base + VGPR int32 byte offset.
#define ASYNC_LD16(ldsAddr, vOff, sBase)                                  \
    asm volatile("global_load_async_to_lds_b128 %0, %1, %2"               \
                 :: "v"(ldsAddr), "v"(vOff), "s"(sBase) : "memory")

#define WAIT_ASYNC0() asm volatile("s_wait_asynccnt 0x0" ::: "memory")

__global__ __launch_bounds__(256)
void router_wmma_f32(const float* __restrict__ x, const float* __restrict__ W,
                     int* __restrict__ onehot, float* __restrict__ topp,
                     float* __restrict__ logits)
{
    __shared__ __align__(16) float sX[2 * XTILE];   // 2 x 34816 B
    __shared__ __align__(16) float sW[2 * WTILE];   // 2 x 17408 B
    __shared__ int sAmax[T_PER_BLOCK];

    const int tid  = threadIdx.x;
    const int wave = tid >> 5;
    const int lane = tid & 31;
    const int l16  = lane & 15;
    const int t0   = blockIdx.x * T_PER_BLOCK;

    // LDS byte addresses (low 32 bits of the shared-aperture flat address)
    const unsigned sxb = (unsigned)(uintptr_t)(&sX[0]);
    const unsigned swb = (unsigned)(uintptr_t)(&sW[0]);

    v8f acc0 = {}, acc1 = {}, acc2 = {}, acc3 = {};

    // ---- async stage of one K-chunk into buffer `buf` ----
    auto stage = [&](int buf, int k0) {
        const unsigned sx_base = sxb + (unsigned)buf * (XTILE * 4u);
        const unsigned sw_base = swb + (unsigned)buf * (WTILE * 4u);
#pragma unroll
        for (int i = 0; i < 8; ++i) {                 // x tile: 128 x 64 f32
            int flat = tid + 256 * i;                 // float4 slots
            int row  = flat >> 4;
            int c4   = (flat & 15) << 2;
            unsigned voff = (unsigned)(t0 + row) * (D_MODEL * 4u) + (unsigned)(k0 + c4) * 4u;
            unsigned loff = sx_base + (unsigned)(row * LDST + c4) * 4u;
            ASYNC_LD16(loff, voff, x);
        }
#pragma unroll
        for (int i = 0; i < 4; ++i) {                 // W tile: 64 x 64 f32
            int flat = tid + 256 * i;
            int row  = flat >> 4;
            int c4   = (flat & 15) << 2;
            unsigned voff = (unsigned)row * (D_MODEL * 4u) + (unsigned)(k0 + c4) * 4u;
            unsigned loff = sw_base + (unsigned)(row * LDST + c4) * 4u;
            ASYNC_LD16(loff, voff, W);
        }
    };

    // prologue: fill buffer 0
    stage(0, 0);
    WAIT_ASYNC0();
    __syncthreads();

    const int arow = (wave << 4) + l16;
    const int koff = (lane >> 4) << 1;                // 0 or 2

    int buf = 0;
    for (int k0 = 0; k0 < D_MODEL; k0 += BK) {
        if (k0 + BK < D_MODEL) stage(buf ^ 1, k0 + BK);   // overlap DMA with WMMA

        const float* bx = &sX[buf * XTILE];
        const float* bw = &sW[buf * WTILE];
        // A 16x4 f32 frag: lanes 0-15 -> M=lane, K=(kk,kk+1); lanes 16-31 -> K=(kk+2,kk+3)
        // B 4x16 f32 frag: lanes 0-15 -> N=lane, same K split.
#pragma unroll
        for (int kk = 0; kk < BK; kk += 4) {
            v2f a  = *(const v2f*)(&bx[arow * LDST + kk + koff]);
            v2f b0 = *(const v2f*)(&bw[(l16     ) * LDST + kk + koff]);
            v2f b1 = *(const v2f*)(&bw[(l16 + 16) * LDST + kk + koff]);
            v2f b2 = *(const v2f*)(&bw[(l16 + 32) * LDST + kk + koff]);
            v2f b3 = *(const v2f*)(&bw[(l16 + 48) * LDST + kk + koff]);
            acc0 = __builtin_amdgcn_wmma_f32_16x16x4_f32(false, a, false, b0, (short)0, acc0, false, false);
            acc1 = __builtin_amdgcn_wmma_f32_16x16x4_f32(false, a, false, b1, (short)0, acc1, false, false);
            acc2 = __builtin_amdgcn_wmma_f32_16x16x4_f32(false, a, false, b2, (short)0, acc2, false, false);
            acc3 = __builtin_amdgcn_wmma_f32_16x16x4_f32(false, a, false, b3, (short)0, acc3, false, false);
        }

        WAIT_ASYNC0();          // next chunk's DMA had the whole compute phase to land
        __syncthreads();
        buf ^= 1;
    }

    // ---- epilogue: dump accumulators to LDS in [token][expert] layout (buffer 0) ----
    {
        // C/D layout: VGPR j holds M=j (lanes 0-15) or M=j+8 (lanes 16-31), N = lane&15
        const int mbase = (wave << 4) + ((lane >> 4) << 3);
#pragma unroll
        for (int j = 0; j < 8; ++j) {
            float* r = &sX[(mbase + j) * LDST];
            r[l16     ] = acc0[j];
            r[l16 + 16] = acc1[j];
            r[l16 + 32] = acc2[j];
            r[l16 + 48] = acc3[j];
        }
    }
    __syncthreads();

    // ---- per-token softmax stats (16 lanes per wave, one token each) ----
    if (lane < 16) {
        const int tl = (wave << 4) + lane;
        const float* r = &sX[tl * LDST];
        float best = r[0]; int bi = 0;
#pragma unroll
        for (int e = 1; e < NUM_EXPERTS; ++e) {
            float v = r[e];
            if (v > best) { best = v; bi = e; }     // first-max semantics like jnp.argmax
        }
        float s = 0.0f;
#pragma unroll
        for (int e = 0; e < NUM_EXPERTS; ++e) s += __expf(r[e] - best);
        topp[t0 + tl] = 1.0f / s;                   // max prob = exp(0)/sum
        sAmax[tl] = bi;
    }
    __syncthreads();

    // ---- coalesced b128 stores: logits + one-hot ----
#pragma unroll
    for (int i = 0; i < 8; ++i) {
        int flat = tid + 256 * i;                   // 0..2047 float4 slots (128 tok x 16)
        int tl   = flat >> 4;
        int c4   = (flat & 15) << 2;
        float4 v = *(const float4*)(&sX[tl * LDST + c4]);
        *(float4*)(logits + (size_t)(t0 + tl) * NUM_EXPERTS + c4) = v;
        int am = sAmax[tl];
        int4 o;
        o.x = (c4     == am);
        o.y = (c4 + 1 == am);
        o.z = (c4 + 2 == am);
        o.w = (c4 + 3 == am);
        *(int4*)(onehot + (size_t)(t0 + tl) * NUM_EXPERTS + c4) = o;
    }
}

extern "C" void kernel_launch(void* const* d_in, const int* in_sizes, int n_in,
                              void* d_out, int out_size, void* d_ws, size_t ws_size,
                              hipStream_t stream) {
    const float* x = (const float*)d_in[0];
    const float* W = (const float*)d_in[1];
    const int num_tokens = in_sizes[0] / D_MODEL;         // 65536

    int*   onehot = (int*)d_out;
    float* topp   = (float*)d_out + (size_t)num_tokens * NUM_EXPERTS;
    float* logits = topp + num_tokens;

    dim3 grid(num_tokens / T_PER_BLOCK);
    dim3 block(256);
    hipLaunchKernelGGL(router_wmma_f32, grid, block, 0, stream, x, W, onehot, topp, logits);
}